// SoftHistogram_77481210020183
// MI455X (gfx1250) — compile-verified
//
#include <hip/hip_runtime.h>
#include <hip/hip_bf16.h>

// Soft histogram, CDNA5 / gfx1250 (wave32).
//
// k(x,b) = sigmoid(x-b) - sigmoid(x-b-1)
//        = C1*u / (1 + C2*u + u^2),  u = exp(-|x - (b+0.5)|)
//   C1 = 2*sinh(0.5), C2 = 2*cosh(0.5).  u in (0,1] -> no overflow; underflow->0 (exact limit).
//
// Kernel 1: bins->threads; 1 v_exp + 1 v_rcp + ~5 VALU per eval; tile staged to LDS via
//           the gfx1250 async path (global_load_async_to_lds_b128 + s_wait_asynccnt);
//           partial histograms merged with global_atomic_add_f32.
// Kernel 2: per-channel normalization; bin-sum via v_wmma_f32_16x16x4_f32 ones-trick
//           (A = ones -> D rows = column sums; ds_swizzle xor-reduce finishes; the
//           32-lane sum double-counts -> 2*S).

#define BINS     256
#define LEN      65536         // elements per channel
#define CHANNELS 12
#define BPC      64            // blocks per channel
#define TILE     (LEN / BPC)   // 1024 elements per block (4 KB LDS)
#define BLOCK    256

typedef __attribute__((ext_vector_type(2))) float v2f;
typedef __attribute__((ext_vector_type(8))) float v8f;
typedef __attribute__((ext_vector_type(4))) int   v4i;

// Address-space-qualified pointer element types for the async-LDS builtin
// (diagnostic showed param0 = "int __vector(4) __device__ *", i.e. v4i in AS1).
typedef __attribute__((address_space(1))) v4i ga_v4i;   // global
typedef __attribute__((address_space(3))) v4i ls_v4i;   // LDS

__global__ __launch_bounds__(BLOCK) void
SoftHistogram_accum(const float* __restrict__ x, float* __restrict__ out) {
    __shared__ float sx[TILE];
    const int ch = blockIdx.y;          // 0..11
    const int t  = threadIdx.x;         // bin id, 0..255

    // Stage this block's 1024 elements into LDS: 16 bytes per lane.
    const float4* src = (const float4*)(x + (size_t)ch * LEN + (size_t)blockIdx.x * TILE);
#if __has_builtin(__builtin_amdgcn_global_load_async_to_lds_b128) && \
    __has_builtin(__builtin_amdgcn_s_wait_asynccnt)
    {
        float4* gsrc = const_cast<float4*>(src) + t;            // strip const
        __builtin_amdgcn_global_load_async_to_lds_b128(
            (ga_v4i*)gsrc,                                       // reinterpret + AS1 cast
            (ls_v4i*)(((float4*)sx) + t),                        // addrspacecast generic->AS3
            /*offset=*/0, /*cpol=*/0);
        __builtin_amdgcn_s_wait_asynccnt(0);
    }
#else
    ((float4*)sx)[t] = src[t];
#endif
    __syncthreads();

    const float center = (float)t + 0.5f;
    const float NL2E = -1.4426950408889634f;   // -log2(e): exp(-|d|) = exp2(NL2E*|d|)
    const float C1   =  1.0421906109874948f;   // 2*sinh(0.5)
    const float C2   =  2.2552519304127614f;   // 2*cosh(0.5)

    // 4 independent accumulators: the loop-carried fma chain must not serialize the
    // TRANS-pipe (v_exp/v_rcp) throughput, which is the real limit (2 trans / eval).
    float acc0 = 0.0f, acc1 = 0.0f, acc2 = 0.0f, acc3 = 0.0f;
    const float4* s4 = (const float4*)sx;
    #pragma unroll 4
    for (int e = 0; e < TILE / 4; ++e) {
        float4 q = s4[e];   // uniform address: one ds_load_b128 broadcast per 4 evals
        {
            float d = q.x - center;
            float u = exp2f(NL2E * fabsf(d));
            float r = __builtin_amdgcn_rcpf(fmaf(u, u, fmaf(u, C2, 1.0f)));
            acc0 = fmaf(C1 * u, r, acc0);
        }
        {
            float d = q.y - center;
            float u = exp2f(NL2E * fabsf(d));
            float r = __builtin_amdgcn_rcpf(fmaf(u, u, fmaf(u, C2, 1.0f)));
            acc1 = fmaf(C1 * u, r, acc1);
        }
        {
            float d = q.z - center;
            float u = exp2f(NL2E * fabsf(d));
            float r = __builtin_amdgcn_rcpf(fmaf(u, u, fmaf(u, C2, 1.0f)));
            acc2 = fmaf(C1 * u, r, acc2);
        }
        {
            float d = q.w - center;
            float u = exp2f(NL2E * fabsf(d));
            float r = __builtin_amdgcn_rcpf(fmaf(u, u, fmaf(u, C2, 1.0f)));
            acc3 = fmaf(C1 * u, r, acc3);
        }
    }
    float acc = (acc0 + acc1) + (acc2 + acc3);
    atomicAdd(&out[ch * BINS + t], acc);   // global_atomic_add_f32, 64 adders per address
}

__global__ __launch_bounds__(32) void
SoftHistogram_normalize(float* __restrict__ out) {
    const int ch   = blockIdx.x;
    const int lane = threadIdx.x;          // 0..31, one wave
    float* h = out + ch * BINS;

    float vals[8];
    #pragma unroll
    for (int i = 0; i < 8; ++i) vals[i] = h[lane * 8 + i];

    // Ones-trick: D[m,n] = sum_k B[k,n] (+C). Four chained WMMAs consume all 256 values
    // (2 B-VGPRs each); every B slot is counted exactly once, so slot mapping is irrelevant.
    v8f c = {};
    v2f a; a.x = 1.0f; a.y = 1.0f;
    #pragma unroll
    for (int jj = 0; jj < 4; ++jj) {
        v2f b; b.x = vals[2 * jj]; b.y = vals[2 * jj + 1];
        c = __builtin_amdgcn_wmma_f32_16x16x4_f32(false, a, false, b,
                                                  (short)0, c, false, false);
    }

    // D replicated across rows: summing c[0] over all 32 lanes double-counts -> 2*S.
    // xor-reduce with ds_swizzle (group-of-32: offset = xor<<10 | and_mask 0x1F).
    float s = c[0];
    s += __int_as_float(__builtin_amdgcn_ds_swizzle(__float_as_int(s), 0x401F)); // xor 16
    s += __int_as_float(__builtin_amdgcn_ds_swizzle(__float_as_int(s), 0x201F)); // xor 8
    s += __int_as_float(__builtin_amdgcn_ds_swizzle(__float_as_int(s), 0x101F)); // xor 4
    s += __int_as_float(__builtin_amdgcn_ds_swizzle(__float_as_int(s), 0x081F)); // xor 2
    s += __int_as_float(__builtin_amdgcn_ds_swizzle(__float_as_int(s), 0x041F)); // xor 1
    float inv = 2.0f * __builtin_amdgcn_rcpf(s);   // 1/S via v_rcp_f32 (no IEEE div sequence)

    #pragma unroll
    for (int i = 0; i < 8; ++i) h[lane * 8 + i] = vals[i] * inv;
}

extern "C" void kernel_launch(void* const* d_in, const int* in_sizes, int n_in,
                              void* d_out, int out_size, void* d_ws, size_t ws_size,
                              hipStream_t stream) {
    (void)in_sizes; (void)n_in; (void)d_ws; (void)ws_size;
    const float* x = (const float*)d_in[0];
    float* out = (float*)d_out;

    // Zero the accumulation target (harness poisons d_out; memset is graph-capturable).
    (void)hipMemsetAsync(out, 0, (size_t)out_size * sizeof(float), stream);

    dim3 grid(BPC, CHANNELS);
    SoftHistogram_accum<<<grid, BLOCK, 0, stream>>>(x, out);
    SoftHistogram_normalize<<<CHANNELS, 32, 0, stream>>>(out);
}